// RPNPostProcessor_73521250173394
// MI455X (gfx1250) — compile-verified
//
#include <hip/hip_runtime.h>
#include <stdint.h>

#define KTOP 4096
#define POST 500
#define NMS_THR 0.7f
#define SORTN 8192

// ---------- helpers ----------
__device__ __forceinline__ uint32_t fkey(float f) {
  uint32_t u = __float_as_uint(f);
  uint32_t mask = (uint32_t)((int32_t)u >> 31) | 0x80000000u;
  return u ^ mask;  // monotonic: bigger float -> bigger key
}
__device__ __forceinline__ float keyToFloat(uint32_t k) {
  uint32_t u = (k & 0x80000000u) ? (k ^ 0x80000000u) : ~k;
  return __uint_as_float(u);
}

// ---------- zero scratch ----------
__global__ void k_zero(uint32_t* p, int n) {
  int i = blockIdx.x * blockDim.x + threadIdx.x;
  if (i < n) p[i] = 0u;
}

// ---------- histogram level 0/1/2 (11+11+10 bits => exact 32-bit threshold) ----------
__global__ void k_hist(const float* __restrict__ obj, uint32_t* __restrict__ hist,
                       const uint32_t* __restrict__ binSel, int level, int B, int A) {
  __shared__ uint32_t lh[2048];
  for (int i = threadIdx.x; i < 2048; i += blockDim.x) lh[i] = 0u;
  __syncthreads();
  int b = blockIdx.y;
  uint32_t bin0 = 0, bin01 = 0;
  if (level >= 1) bin0 = binSel[b * 4 + 0];
  if (level >= 2) bin01 = (binSel[b * 4 + 0] << 11) | binSel[b * 4 + 1];
  const float* ob = obj + (size_t)b * A;
  for (int i = blockIdx.x * blockDim.x + threadIdx.x; i < A; i += blockDim.x * gridDim.x) {
    uint32_t key = fkey(ob[i]);
    if (level == 0) {
      atomicAdd(&lh[key >> 21], 1u);
    } else if (level == 1) {
      if ((key >> 21) == bin0) atomicAdd(&lh[(key >> 10) & 0x7FFu], 1u);
    } else {
      if ((key >> 10) == bin01) atomicAdd(&lh[key & 0x3FFu], 1u);
    }
  }
  __syncthreads();
  uint32_t* gh = hist + ((size_t)level * B + b) * 2048;
  for (int i = threadIdx.x; i < 2048; i += blockDim.x)
    if (lh[i]) atomicAdd(&gh[i], lh[i]);
}

// ---------- pick the threshold bin at each level ----------
__global__ void k_pick(const uint32_t* __restrict__ hist, uint32_t* __restrict__ binSel,
                       uint32_t* __restrict__ countAbove, uint32_t* __restrict__ tkey,
                       uint32_t* __restrict__ counters, int level, int B) {
  int b = threadIdx.x;
  if (b >= B) return;
  const uint32_t* h = hist + ((size_t)level * B + b) * 2048;
  uint32_t prev = (level == 0) ? 0u : countAbove[b];
  int bins = (level == 2) ? 1024 : 2048;
  uint32_t cum = prev;
  int sel = 0;
  for (int c = bins - 1; c >= 0; --c) {
    uint32_t nc = cum + h[c];
    if (nc >= (uint32_t)KTOP) { sel = c; break; }
    cum = nc;
  }
  binSel[b * 4 + level] = (uint32_t)sel;
  countAbove[b] = cum;  // strictly above selected bin
  if (level == 2) {
    tkey[b] = (binSel[b * 4 + 0] << 21) | (binSel[b * 4 + 1] << 10) | (uint32_t)sel;
    counters[b * 2 + 0] = 0u;
    counters[b * 2 + 1] = 0u;
  }
}

// ---------- compact keys > T and keys == T ----------
__global__ void k_compact(const float* __restrict__ obj, const uint32_t* __restrict__ tkey,
                          uint32_t* __restrict__ counters, uint32_t* __restrict__ gtKey,
                          uint32_t* __restrict__ gtIdx, uint32_t* __restrict__ eqIdx,
                          int B, int A) {
  int b = blockIdx.y;
  uint32_t T = tkey[b];
  const float* ob = obj + (size_t)b * A;
  for (int i = blockIdx.x * blockDim.x + threadIdx.x; i < A; i += blockDim.x * gridDim.x) {
    uint32_t key = fkey(ob[i]);
    if (key > T) {
      uint32_t p = atomicAdd(&counters[b * 2 + 0], 1u);  // guaranteed < KTOP
      gtKey[(size_t)b * KTOP + p] = key;
      gtIdx[(size_t)b * KTOP + p] = (uint32_t)i;
    } else if (key == T) {
      uint32_t p = atomicAdd(&counters[b * 2 + 1], 1u);
      if (p < (uint32_t)KTOP) eqIdx[(size_t)b * KTOP + p] = (uint32_t)i;
    }
  }
}

// ---------- per-batch bitonic sort of (key, ~idx) in 64KB LDS ----------
__global__ void __launch_bounds__(1024) k_sortk(
    const uint32_t* __restrict__ counters, const uint32_t* __restrict__ gtKey,
    const uint32_t* __restrict__ gtIdx, const uint32_t* __restrict__ eqIdx,
    const uint32_t* __restrict__ tkey, uint32_t* __restrict__ topIdx,
    float* __restrict__ topScore, int B) {
  __shared__ unsigned long long arr[SORTN];
  int b = blockIdx.x;
  uint32_t nGT = counters[b * 2 + 0];
  uint32_t cEQ = counters[b * 2 + 1];
  uint32_t nEQ = cEQ < (uint32_t)KTOP ? cEQ : (uint32_t)KTOP;
  uint32_t T = tkey[b];
  for (int i = threadIdx.x; i < SORTN; i += blockDim.x) {
    unsigned long long v = 0ull;  // pad: sorts last
    if ((uint32_t)i < nGT)
      v = ((unsigned long long)gtKey[(size_t)b * KTOP + i] << 32) |
          (uint32_t)(~gtIdx[(size_t)b * KTOP + i]);
    else if ((uint32_t)i < nGT + nEQ)
      v = ((unsigned long long)T << 32) | (uint32_t)(~eqIdx[(size_t)b * KTOP + (i - nGT)]);
    arr[i] = v;
  }
  __syncthreads();
  // bitonic sort, descending (key desc, then idx asc via ~idx)
  for (int k2 = 2; k2 <= SORTN; k2 <<= 1) {
    for (int j = k2 >> 1; j > 0; j >>= 1) {
      for (int i = threadIdx.x; i < SORTN; i += blockDim.x) {
        int ixj = i ^ j;
        if (ixj > i) {
          unsigned long long a = arr[i], c = arr[ixj];
          bool descRegion = ((i & k2) == 0);
          bool sw = descRegion ? (a < c) : (a > c);
          if (sw) { arr[i] = c; arr[ixj] = a; }
        }
      }
      __syncthreads();
    }
  }
  for (int r = threadIdx.x; r < KTOP; r += blockDim.x) {
    unsigned long long v = arr[r];
    uint32_t idx = ~(uint32_t)v;
    float o = keyToFloat((uint32_t)(v >> 32));
    topIdx[(size_t)b * KTOP + r] = idx;
    topScore[(size_t)b * KTOP + r] = 1.0f / (1.0f + expf(-o));
  }
}

// ---------- gather + decode boxes, precompute BEV rects/areas ----------
__global__ void k_decode(const float* __restrict__ anchors, const float* __restrict__ reg,
                         const uint32_t* __restrict__ topIdx, float* __restrict__ boxes,
                         float4* __restrict__ rect4, float* __restrict__ areaArr,
                         int B, int A) {
  int g = blockIdx.x * blockDim.x + threadIdx.x;
  if (g >= B * KTOP) return;
  int b = g >> 12;
  uint32_t idx = topIdx[g];
  const float* an = anchors + ((size_t)b * A + idx) * 7;
  const float* rg = reg + ((size_t)b * A + idx) * 7;
  float xa = an[0], ya = an[1], za = an[2], wa = an[3], la = an[4], ha = an[5], ra = an[6];
  float diag = sqrtf(wa * wa + la * la);
  float xg = rg[0] * diag + xa;
  float yg = rg[1] * diag + ya;
  float zg = rg[2] * ha + za;
  float wg = expf(rg[3]) * wa;
  float lg = expf(rg[4]) * la;
  float hg = expf(rg[5]) * ha;
  float rr = rg[6] + ra;
  float* bx = boxes + (size_t)g * 7;
  bx[0] = xg; bx[1] = yg; bx[2] = zg; bx[3] = wg; bx[4] = lg; bx[5] = hg; bx[6] = rr;
  rect4[g] = make_float4(xg - wg * 0.5f, yg - lg * 0.5f, xg + wg * 0.5f, yg + lg * 0.5f);
  areaArr[g] = wg * lg;
}

// ---------- suppression bitmask: 64x64 tiles, j-tile staged to LDS (async on gfx1250) ----------
__global__ void __launch_bounds__(64) k_mask(const float4* __restrict__ rect4,
                                             const float* __restrict__ areaArr,
                                             uint32_t* __restrict__ mask, int B) {
  __shared__ float4 srect[64];
  __shared__ float sarea[64];
  int b = blockIdx.z;
  int it = blockIdx.x, jt = blockIdx.y;
  int t = threadIdx.x;
  size_t base = (size_t)b * KTOP;
  const float4* gr = rect4 + base + jt * 64 + t;
  const float* ga = areaArr + base + jt * 64 + t;
#if defined(__gfx1250__)
  {
    uint32_t lr = (uint32_t)(uintptr_t)(&srect[t]);  // low 32 bits of flat addr = LDS offset
    uint32_t la = (uint32_t)(uintptr_t)(&sarea[t]);
    asm volatile("global_load_async_to_lds_b128 %0, %1, off" ::"v"(lr), "v"(gr) : "memory");
    asm volatile("global_load_async_to_lds_b32 %0, %1, off" ::"v"(la), "v"(ga) : "memory");
    asm volatile("s_wait_asynccnt 0" ::: "memory");
  }
#else
  srect[t] = *gr;
  sarea[t] = *ga;
#endif
  __syncthreads();
  int i = it * 64 + t;
  float4 ri = rect4[base + i];
  float ai = areaArr[base + i];
  uint32_t m0 = 0, m1 = 0;
#pragma unroll 4
  for (int jj = 0; jj < 64; ++jj) {
    int j = jt * 64 + jj;
    float4 rj = srect[jj];
    float iw = fminf(ri.z, rj.z) - fmaxf(ri.x, rj.x);
    float ih = fminf(ri.w, rj.w) - fmaxf(ri.y, rj.y);
    iw = fmaxf(iw, 0.0f);
    ih = fmaxf(ih, 0.0f);
    float inter = iw * ih;
    float iou = inter / (ai + sarea[jj] - inter + 1e-6f);
    uint32_t sup = (iou > NMS_THR) && (j > i);
    if (jj < 32) m0 |= sup << jj; else m1 |= sup << (jj - 32);
  }
  uint32_t* row = mask + ((size_t)b * KTOP + i) * (KTOP / 32);
  row[jt * 2 + 0] = m0;
  row[jt * 2 + 1] = m1;
}

// ---------- chunked sequential greedy scan (32 rows/chunk staged in LDS) ----------
__global__ void __launch_bounds__(128) k_nmsscan(const uint32_t* __restrict__ mask,
                                                 uint32_t* __restrict__ keepW, int B) {
  __shared__ uint32_t tile[32 * 128];
  __shared__ uint32_t bcast;      // remv word of current chunk
  __shared__ uint32_t chunkKeep;  // kept bits of current chunk
  int b = blockIdx.x;
  int t = threadIdx.x;
  uint32_t remv = 0;
  const uint32_t* mb = mask + (size_t)b * KTOP * (KTOP / 32);
  for (int c = 0; c < 128; ++c) {
#if defined(__gfx1250__)
    for (int i2 = 0; i2 < 32; ++i2) {
      uint32_t l = (uint32_t)(uintptr_t)(&tile[i2 * 128 + t]);
      const uint32_t* g = &mb[(size_t)(c * 32 + i2) * 128 + t];
      asm volatile("global_load_async_to_lds_b32 %0, %1, off" ::"v"(l), "v"(g) : "memory");
    }
    asm volatile("s_wait_asynccnt 0" ::: "memory");
#else
    for (int i2 = 0; i2 < 32; ++i2)
      tile[i2 * 128 + t] = mb[(size_t)(c * 32 + i2) * 128 + t];
#endif
    if (t == c) bcast = remv;
    __syncthreads();
    if (t == 0) {
      uint32_t r = bcast, km = 0;
      for (int i2 = 0; i2 < 32; ++i2) {
        if (!((r >> i2) & 1u)) { km |= 1u << i2; r |= tile[i2 * 128 + c]; }
      }
      chunkKeep = km;
    }
    __syncthreads();
    uint32_t km = chunkKeep;
    for (int i2 = 0; i2 < 32; ++i2)
      if ((km >> i2) & 1u) remv |= tile[i2 * 128 + t];
    __syncthreads();
  }
  keepW[b * 128 + t] = ~remv;
}

// ---------- rank (kept-in-order, then non-kept ascending) and emit top-500 ----------
__global__ void __launch_bounds__(128) k_output(const uint32_t* __restrict__ keepW,
                                                const float* __restrict__ boxes,
                                                const float* __restrict__ topScore,
                                                float* __restrict__ out, int B) {
  __shared__ uint32_t scanBuf[128];
  int b = blockIdx.x, t = threadIdx.x;
  uint32_t kw = keepW[b * 128 + t];
  uint32_t pc = __popc(kw);
  scanBuf[t] = pc;
  __syncthreads();
  for (int off = 1; off < 128; off <<= 1) {
    uint32_t v = scanBuf[t];
    uint32_t add = (t >= off) ? scanBuf[t - off] : 0u;
    __syncthreads();
    scanBuf[t] = v + add;
    __syncthreads();
  }
  uint32_t totalKept = scanBuf[127];
  uint32_t keptBase = scanBuf[t] - pc;  // exclusive prefix of kept
  uint32_t nonBase = totalKept + (uint32_t)(t * 32) - keptBase;
  for (int i2 = 0; i2 < 32; ++i2) {
    int r = t * 32 + i2;
    bool kept = (kw >> i2) & 1u;
    uint32_t rank = kept ? keptBase : nonBase;
    if (kept) ++keptBase; else ++nonBase;
    if (rank < (uint32_t)POST) {
      const float* bx = boxes + ((size_t)b * KTOP + r) * 7;
      float* o = out + ((size_t)b * POST + rank) * 8;
      o[0] = bx[0]; o[1] = bx[1]; o[2] = bx[2]; o[3] = bx[3];
      o[4] = bx[4]; o[5] = bx[5]; o[6] = bx[6];
      o[7] = topScore[(size_t)b * KTOP + r];
    }
  }
}

extern "C" void kernel_launch(void* const* d_in, const int* in_sizes, int n_in,
                              void* d_out, int out_size, void* d_ws, size_t ws_size,
                              hipStream_t stream) {
  (void)n_in; (void)out_size; (void)ws_size;
  const float* anchors = (const float*)d_in[0];
  const float* obj     = (const float*)d_in[1];
  const float* reg     = (const float*)d_in[2];
  float* out = (float*)d_out;
  const int A = 500000;
  int B = in_sizes[1] / A;
  if (B < 1) B = 1;

  uint8_t* ws = (uint8_t*)d_ws;
  size_t off = 0;
  auto alloc = [&](size_t bytes) -> void* {
    void* p = ws + off;
    off = (off + bytes + 255) & ~(size_t)255;
    return p;
  };
  uint32_t* hist       = (uint32_t*)alloc((size_t)3 * B * 2048 * 4);
  uint32_t* binSel     = (uint32_t*)alloc((size_t)B * 4 * 4);
  uint32_t* countAbove = (uint32_t*)alloc((size_t)B * 4);
  uint32_t* tkey       = (uint32_t*)alloc((size_t)B * 4);
  uint32_t* counters   = (uint32_t*)alloc((size_t)B * 2 * 4);
  uint32_t* gtKey      = (uint32_t*)alloc((size_t)B * KTOP * 4);
  uint32_t* gtIdx      = (uint32_t*)alloc((size_t)B * KTOP * 4);
  uint32_t* eqIdx      = (uint32_t*)alloc((size_t)B * KTOP * 4);
  uint32_t* topIdx     = (uint32_t*)alloc((size_t)B * KTOP * 4);
  float*    topScore   = (float*)alloc((size_t)B * KTOP * 4);
  float*    boxes      = (float*)alloc((size_t)B * KTOP * 7 * 4);
  float4*   rect4      = (float4*)alloc((size_t)B * KTOP * 16);
  float*    areaArr    = (float*)alloc((size_t)B * KTOP * 4);
  uint32_t* mask       = (uint32_t*)alloc((size_t)B * KTOP * (KTOP / 32) * 4);
  uint32_t* keepW      = (uint32_t*)alloc((size_t)B * (KTOP / 32) * 4);

  int zn = 3 * B * 2048;
  k_zero<<<(zn + 255) / 256, 256, 0, stream>>>(hist, zn);
  for (int level = 0; level < 3; ++level) {
    k_hist<<<dim3(128, B), 256, 0, stream>>>(obj, hist, binSel, level, B, A);
    k_pick<<<1, 32, 0, stream>>>(hist, binSel, countAbove, tkey, counters, level, B);
  }
  k_compact<<<dim3(256, B), 256, 0, stream>>>(obj, tkey, counters, gtKey, gtIdx, eqIdx, B, A);
  k_sortk<<<B, 1024, 0, stream>>>(counters, gtKey, gtIdx, eqIdx, tkey, topIdx, topScore, B);
  k_decode<<<(B * KTOP + 255) / 256, 256, 0, stream>>>(anchors, reg, topIdx, boxes, rect4,
                                                       areaArr, B, A);
  k_mask<<<dim3(KTOP / 64, KTOP / 64, B), 64, 0, stream>>>(rect4, areaArr, mask, B);
  k_nmsscan<<<B, 128, 0, stream>>>(mask, keepW, B);
  k_output<<<B, 128, 0, stream>>>(keepW, boxes, topScore, out, B);
}